// TrainableActivationKeras_23175643529854
// MI455X (gfx1250) — compile-verified
//
#include <hip/hip_runtime.h>
#include <math.h>

// Gaussian-RBF trainable activation for MI455X (gfx1250, wave32).
//   out[e] = sum_j w[c,j] * exp(-112.5*(x[e]-mu_j)^2),  mu_j = -1 + j/15, c = e & 63
//
// sigma = 1/15 => Gaussians are so narrow that only the +/-7 centers nearest x
// contribute above fp32 noise (term_d <= e^{-0.5 d(d-1)}; d=7 -> e^{-21}).
// Anchor at nearest center j0 and factorize each term as E * r^d * e^{-d^2/2}:
//   t = 15*(x+1), j0 = clamp(rint(t),0,30), u = t - j0
//   E = exp(-u^2/2), r = exp(u)
// -> 3 v_exp_f32 per element instead of 31. Powers built incrementally with the
// e^{-d^2/2} constants folded in so nothing overflows (u clamped to +/-15 =>
// max exponent e^{80.5} < FLT_MAX; E underflows to 0 exactly where the fp32
// reference underflows).

#define NW     31
#define NC     64
#define HALO   7
#define WIN    15              // 2*HALO + 1
#define ROWS   (NW + 2*HALO)   // 45 rows; rows outside [7,37] are zero padding

typedef __attribute__((address_space(1))) int g_int;
typedef __attribute__((address_space(3))) int l_int;

__device__ __forceinline__ float fast_exp2(float a) {
#if __has_builtin(__builtin_amdgcn_exp2f)
    return __builtin_amdgcn_exp2f(a);   // raw v_exp_f32
#else
    return exp2f(a);
#endif
}

__device__ __forceinline__ float rbf_eval(float x, const float* __restrict__ Wcol) {
    // Wcol = Wlds + c  (per-lane channel column, row stride NC floats = 256 B)
    const float LOG2E = 1.4426950408889634f;
    float t   = __builtin_fmaf(x, 15.0f, 15.0f);     // grid coords, centers at 0..30
    float j0f = rintf(t);
    j0f = fminf(fmaxf(j0f, 0.0f), 30.0f);
    float u = t - j0f;                               // 15*(x - mu_j0)
    u = fminf(fmaxf(u, -15.0f), 15.0f);              // keep powers finite for wild x
    int j0 = (int)j0f;

    // three transcendentals total
    float a  = u * LOG2E;
    float E  = fast_exp2(u * a * -0.5f);             // exp(-u^2/2)
    float r  = fast_exp2(a);                         // exp(+u)
    float ri = fast_exp2(-a);                        // exp(-u)

    // 15 LDS loads; constant 256 B row offsets -> ds_load_2addr_stride64 pairs
    const float* Wrow = Wcol + j0 * NC;              // row j0+k holds weight j = j0+k-7
    float w[WIN];
#pragma unroll
    for (int k = 0; k < WIN; ++k) w[k] = Wrow[k * NC];

    // step[d-1] = e^{-(2d-1)/2}  so  q_d = q_{d-1} * r * step = r^d e^{-d^2/2}
    const float STEP[HALO] = {0.60653066f,  0.22313017f,   0.082084999f, 0.030197383f,
                              0.011108997f, 0.0040867714f, 0.0015034392f};
    float acc = w[HALO];                             // d = 0 (coef 1)
    float qp = 1.0f, qn = 1.0f;
#pragma unroll
    for (int d = 1; d <= HALO; ++d) {
        qp = qp * r;  qp = qp * STEP[d - 1];
        qn = qn * ri; qn = qn * STEP[d - 1];
        acc = __builtin_fmaf(w[HALO + d], qp, acc);
        acc = __builtin_fmaf(w[HALO - d], qn, acc);
    }
    return acc * E;
}

__global__ void __launch_bounds__(256)
rbf_act_kernel(const float* __restrict__ x, const float* __restrict__ wgt,
               float* __restrict__ out, int n) {
    __shared__ float Wlds[ROWS * NC];
    const int tid = threadIdx.x;

    // 1) zero the halo rows (rows 0..6 and 38..44)
    for (int idx = tid; idx < ROWS * NC; idx += blockDim.x) {
        int ri = idx >> 6;
        if (ri < HALO || ri >= HALO + NW) Wlds[idx] = 0.0f;
    }

    // 2) stage weights global->LDS with transpose: wgt is (C=64, Nw=31) row-major,
    //    LDS wants [j+7][c]. Global side reads linearly (coalesced); the scatter
    //    happens on the LDS destination address. Uses the gfx1250 async
    //    global->LDS path (ASYNCcnt), no VGPR round-trip.
#if __has_builtin(__builtin_amdgcn_global_load_async_to_lds_b32)
    for (int idx = tid; idx < NC * NW; idx += blockDim.x) {
        int c = idx / NW;
        int j = idx - c * NW;
        int tgt = (j + HALO) * NC + c;
        __builtin_amdgcn_global_load_async_to_lds_b32(
            (g_int*)(wgt + idx),
            (l_int*)(&Wlds[tgt]),
            /*offset=*/0, /*cpol=*/0);
    }
#if __has_builtin(__builtin_amdgcn_s_wait_asynccnt)
    __builtin_amdgcn_s_wait_asynccnt(0);
#else
    asm volatile("s_wait_asynccnt 0x0" ::: "memory");
#endif
#else
    for (int idx = tid; idx < NC * NW; idx += blockDim.x) {
        int c = idx / NW;
        int j = idx - c * NW;
        Wlds[(j + HALO) * NC + c] = wgt[idx];
    }
#endif
    __syncthreads();

    const int gsz = gridDim.x * blockDim.x;          // multiple of 64 (256 threads/blk)
    const int i0  = blockIdx.x * blockDim.x + tid;
    // loop stride 2*gsz is a multiple of 64 -> channel is loop-invariant; hoist it
    const float* Wcol0 = Wlds + (i0 & 63);
    const float* Wcol1 = Wlds + ((i0 + gsz) & 63);

    for (int i = i0; i < n; i += 2 * gsz) {
        int i2 = i + gsz;
        int inext = i + 2 * gsz;
        if (inext < n) __builtin_prefetch(x + inext, 0, 0);   // global_prefetch_b8

        float x0 = __builtin_nontemporal_load(x + i);         // streamed: skip L2 pollution
        float y0 = rbf_eval(x0, Wcol0);
        __builtin_nontemporal_store(y0, out + i);

        if (i2 < n) {
            float x1 = __builtin_nontemporal_load(x + i2);
            float y1 = rbf_eval(x1, Wcol1);
            __builtin_nontemporal_store(y1, out + i2);
        }
    }
}

extern "C" void kernel_launch(void* const* d_in, const int* in_sizes, int n_in,
                              void* d_out, int out_size, void* d_ws, size_t ws_size,
                              hipStream_t stream) {
    (void)n_in; (void)d_ws; (void)ws_size; (void)out_size;
    const float* x   = (const float*)d_in[0];   // (8,256,256,64) fp32
    const float* wgt = (const float*)d_in[1];   // (64,31) fp32
    float* out = (float*)d_out;
    int n = in_sizes[0];

    const int threads = 256;                    // 8 wave32 per block
    const int per_blk = threads * 2;            // 2 elements / thread / iter
    long long nb = ((long long)n + per_blk - 1) / per_blk;
    int blocks = (nb > 8192) ? 8192 : (int)(nb < 1 ? 1 : nb);

    rbf_act_kernel<<<blocks, threads, 0, stream>>>(x, wgt, out, n);
}